// MultiSetRNN_49555332661352
// MI455X (gfx1250) — compile-verified
//
#include <hip/hip_runtime.h>

// ---------------- problem constants ----------------
#define NUM_SETS 8
#define IN_SZ    256
#define HID      256
#define G4       1024   // 4*HID, gate order i,f,g,o
#define LATENT   32
#define BB       128
#define TT       512

// LDS row stride for all TDM-loaded tiles: 256 floats payload + 4 floats pad
// (TDM pad_interval = 256 DWORDs, pad_amount = 4 DWORDs). 260 mod 64 = 4 ->
// the 16-lane column reads of WMMA fragments hit 16 distinct LDS banks.
#define RP 260

typedef float v2f  __attribute__((ext_vector_type(2)));
typedef float v8f  __attribute__((ext_vector_type(8)));
typedef unsigned u32x4 __attribute__((ext_vector_type(4)));
typedef int   i32x4 __attribute__((ext_vector_type(4)));
typedef int   i32x8 __attribute__((ext_vector_type(8)));

// D(16x16,f32) += A(16x4,f32) * B(4x16,f32)
// A per lane: M = lane&15, half = lane>>4 -> holds A[M][k0+2*half +{0,1}]
// B per lane: N = lane&15, half          -> holds B[k0+2*half +{0,1}][N]
// C/D per lane: VGPR r -> M = r + 8*half, N = lane&15
__device__ __forceinline__ v8f wmma_f32(v2f a, v2f b, v8f c) {
    return __builtin_amdgcn_wmma_f32_16x16x4_f32(
        false, a, false, b, (short)0, c, false, false);
}

__device__ __forceinline__ float sigmoid_(float x) {
    return 1.0f / (1.0f + __expf(-x));
}
__device__ __forceinline__ float tanh_(float x) {
    float e = __expf(2.0f * x);         // saturates correctly at +/-inf
    return 1.0f - 2.0f / (e + 1.0f);
}

// generic pointer -> LDS byte address (low 32 bits of the flat address are
// the LDS offset; the shared aperture only occupies addr[63:32])
__device__ __forceinline__ unsigned lds_addr_of(const void* p) {
    return (unsigned)(uintptr_t)p;
}

// Tensor Data Mover: DMA a 2-D tile (tile_d0 elems x tile_d1 rows of f32,
// global row stride stride0 elems) into LDS at lds_byte_addr, inserting
// 4 DWORDs of padding after every 256 DWORDs (-> LDS row stride RP floats).
// Descriptor layout per CDNA5 ISA sec 8.3/8.4 (D# groups 0/1; 2D -> groups
// 2/3 zero). Tracked by TENSORcnt.
__device__ __forceinline__ void tdm_load_2d(
    unsigned lds_byte_addr, const void* gptr,
    unsigned tile_d0, unsigned tile_d1,
    unsigned tensor_d0, unsigned tensor_d1,
    unsigned long long stride0)
{
    unsigned long long ga = (unsigned long long)(uintptr_t)gptr;
    u32x4 g0;
    g0[0] = 1u;                                        // count=1 (valid, user)
    g0[1] = lds_byte_addr;                             // lds_addr
    g0[2] = (unsigned)(ga & 0xffffffffu);              // global_addr[31:0]
    g0[3] = (unsigned)((ga >> 32) & 0x01ffffffu)       // global_addr[56:32]
          | 0x80000000u;                               // type=2 ("image")
    i32x8 g1;
    g1[0] = (int)((2u << 16)        // data_size = 4 bytes
                | (1u << 20)        // pad_enable
                | (7u << 22)        // pad_interval: every 256 DWORDs
                | (3u << 25));      // pad_amount: 4 DWORDs
    g1[1] = (int)((tensor_d0 & 0xffffu) << 16);
    g1[2] = (int)(((tensor_d0 >> 16) & 0xffffu) | ((tensor_d1 & 0xffffu) << 16));
    g1[3] = (int)(((tensor_d1 >> 16) & 0xffffu) | ((tile_d0 & 0xffffu) << 16));
    g1[4] = (int)(tile_d1 & 0xffffu);                  // tile_dim1; tile_dim2=0
    g1[5] = (int)(unsigned)(stride0 & 0xffffffffu);    // tensor_dim0_stride
    g1[6] = (int)((stride0 >> 32) & 0xffffu);
    g1[7] = 0;
    i32x4 gz = {0, 0, 0, 0};
#if defined(__clang_major__) && __clang_major__ >= 23
    i32x8 gz8 = {0, 0, 0, 0, 0, 0, 0, 0};
    __builtin_amdgcn_tensor_load_to_lds(g0, g1, gz, gz, gz8, 16 /*scope=DEV*/);
#else
    __builtin_amdgcn_tensor_load_to_lds(g0, g1, gz, gz, 16 /*scope=DEV*/);
#endif
}

// ============================================================
// Kernel 1: emb[b,t,h] = sum_n x[b,t,n] * W_enc[set(b),h,n] + b_enc[set(b),h]
// grid = (T/16, B), block = 256 (8 waves); wave w -> h cols [32w, 32w+32)
// ============================================================
__global__ __launch_bounds__(256) void encoder_kernel(
    const float* __restrict__ x, const int* __restrict__ dsi,
    const float* __restrict__ W_enc, const float* __restrict__ b_enc,
    float* __restrict__ emb)
{
    const int lane = threadIdx.x & 31;
    const int wave = threadIdx.x >> 5;
    const int half = lane >> 4;
    const int l16  = lane & 15;
    const int t0   = blockIdx.x * 16;
    const int b    = blockIdx.y;
    const int set  = dsi[b];

    const float* Arow = x + (size_t)(b * TT + t0 + l16) * IN_SZ + 2 * half;
    const float* W0 = W_enc + ((size_t)set * HID + wave * 32 + l16) * IN_SZ + 2 * half;
    const float* W1 = W0 + (size_t)16 * IN_SZ;

    v8f z = {};
    v8f acc0 = z, acc1 = z;
    #pragma unroll 4
    for (int k0 = 0; k0 < IN_SZ; k0 += 4) {
        v2f a  = *(const v2f*)(Arow + k0);
        v2f b0 = *(const v2f*)(W0 + k0);
        v2f b1 = *(const v2f*)(W1 + k0);
        acc0 = wmma_f32(a, b0, acc0);
        acc1 = wmma_f32(a, b1, acc1);
    }
    const int h0 = wave * 32 + l16;
    const float bias0 = b_enc[set * HID + h0];
    const float bias1 = b_enc[set * HID + h0 + 16];
    #pragma unroll
    for (int r = 0; r < 8; ++r) {
        int m = r + 8 * half;
        size_t o = (size_t)(b * TT + t0 + m) * HID;
        emb[o + h0]      = acc0[r] + bias0;
        emb[o + h0 + 16] = acc1[r] + bias1;
    }
}

// ============================================================
// Kernel 2: gate-parallel batched bidirectional LSTM.
// grid = (8 gate-slices, 8 sample-tiles, 2 dirs) = 128 WGs, block = 256.
// Gate-slice group g owns gate columns {q*256 + g*32 + j : q<4, j<32}
// (i,f,g,o for hidden slice [32g, 32g+32)): its 2x128x256 f32 weight slice
// (256 KB) is TDM-loaded once and stays LDS-RESIDENT (320 KB WGP LDS).
// Per step: TDM stages emb_t tile + broadcast h_prev into LDS
// (s_wait_tensorcnt), each of 8 waves computes one 16x16 gate tile
// (K=512 via two 256 halves, 128 WMMAs, all operands in LDS), fused gate
// math updates c (regs) and h (32 hidden cols), h broadcast via
// double-buffered global buffer + per-group atomic barrier (8 WGs/group).
// ============================================================
__global__ __launch_bounds__(256) void lstm_kernel(
    const float* __restrict__ emb,
    const float* __restrict__ Wih_f, const float* __restrict__ Whh_f, const float* __restrict__ b_f,
    const float* __restrict__ Wih_b, const float* __restrict__ Whh_b, const float* __restrict__ b_b,
    float* __restrict__ hs_f, float* __restrict__ hs_b,
    unsigned* __restrict__ barriers, float* __restrict__ hglob)
{
    __shared__ float wih_lds[128 * RP];   // 130 KB  Wih slice
    __shared__ float whh_lds[128 * RP];   // 130 KB  Whh slice
    __shared__ float e_lds[16 * RP];      //  16 KB  emb_t tile
    __shared__ float h_lds[16 * RP];      //  16 KB  h_prev tile
    __shared__ float g_stage[16 * 128];   //   8 KB  gate pre-activations

    const int g    = blockIdx.x;          // gate-slice group
    const int tile = blockIdx.y;          // sample tile
    const int dir  = blockIdx.z;
    const int s0   = tile * 16;
    const int tid  = threadIdx.x;
    const int lane = tid & 31;
    const int wave = tid >> 5;
    const int half = lane >> 4;
    const int l16  = lane & 15;

    const float* Wih = dir ? Wih_b : Wih_f;
    const float* Whh = dir ? Whh_b : Whh_f;
    const float* bg  = dir ? b_b  : b_f;
    float*       hs  = dir ? hs_b : hs_f;

    const int group = dir * 8 + tile;                 // 16 independent groups
    unsigned* ctr   = barriers + group * 32;          // own cacheline
    float*    hgl   = hglob + (size_t)group * 2 * (16 * HID); // 2 phase bufs

    // ---- one-time weight staging via TDM: 8 x (32 rows x 256 f32) ----
    if (wave == 0) {
        const unsigned wih_a = lds_addr_of(wih_lds);
        const unsigned whh_a = lds_addr_of(whh_lds);
        #pragma unroll
        for (int q = 0; q < 4; ++q) {
            int n0 = (q << 8) + g * 32;               // first gate row of block
            tdm_load_2d(wih_a + (unsigned)(q * 32 * RP * 4),
                        Wih + (size_t)n0 * HID, 256, 32, 256, 32, 256ull);
            tdm_load_2d(whh_a + (unsigned)(q * 32 * RP * 4),
                        Whh + (size_t)n0 * HID, 256, 32, 256, 32, 256ull);
        }
    }

    // biases for this thread's hidden column jp = tid&31 (2 samples/thread)
    const int jp   = tid & 31;
    const int gcol = g * 32 + jp;
    const float bi  = bg[gcol];
    const float bfv = bg[HID + gcol];
    const float bgg = bg[2 * HID + gcol];
    const float bo  = bg[3 * HID + gcol];
    float c0 = 0.0f, c1 = 0.0f;                       // c for s=tid>>5 and +8

    // per-wave fragment pointers (all LDS)
    const float* Ae = e_lds + l16 * RP + 2 * half;
    const float* Ah = h_lds + l16 * RP + 2 * half;
    const float* Bi = wih_lds + (wave * 16 + l16) * RP + 2 * half;
    const float* Bh = whh_lds + (wave * 16 + l16) * RP + 2 * half;

    const unsigned e_a = lds_addr_of(e_lds);
    const unsigned h_a = lds_addr_of(h_lds);

    __builtin_amdgcn_s_wait_tensorcnt(0);
    __syncthreads();

    for (int tt = 0; tt < TT; ++tt) {
        const int t = dir ? (TT - 1 - tt) : tt;

        // ---- stage A = emb_t tile + h_prev broadcast via TDM ----
        if (wave == 0) {
            tdm_load_2d(e_a, emb + (size_t)(s0 * TT + t) * HID,
                        256, 16, 256, 16, (unsigned long long)TT * HID);
            if (tt > 0)
                tdm_load_2d(h_a, hgl + ((tt - 1) & 1) * (16 * HID),
                            256, 16, 256, 16, 256ull);
        }
        if (tt == 0) {
            for (int i = tid; i < 4096; i += 256)
                h_lds[(i >> 8) * RP + (i & 255)] = 0.0f;
        }
        // warm GL2 for the next step's emb tile while the DMA runs
        if (tt + 1 < TT) {
            int tn = dir ? (TT - 2 - tt) : (tt + 1);
            const float* pf = emb + (size_t)((s0 + (tid >> 4)) * TT + tn) * HID
                              + ((tid & 15) << 4);
            __builtin_prefetch(pf, 0, 1);
        }
        __builtin_amdgcn_s_wait_tensorcnt(0);
        __syncthreads();

        // ---- GEMM: wave = one 16x16 gate tile, K = 256 + 256, all LDS ----
        v8f acc = {};
        #pragma unroll 8
        for (int k0 = 0; k0 < HID; k0 += 4) {
            v2f a  = *(const v2f*)(Ae + k0);
            v2f bv = *(const v2f*)(Bi + k0);
            acc = wmma_f32(a, bv, acc);
        }
        #pragma unroll 8
        for (int k0 = 0; k0 < HID; k0 += 4) {
            v2f a  = *(const v2f*)(Ah + k0);
            v2f bv = *(const v2f*)(Bh + k0);
            acc = wmma_f32(a, bv, acc);
        }
        {
            int nl = wave * 16 + l16;
            #pragma unroll
            for (int r = 0; r < 8; ++r)
                g_stage[(r + 8 * half) * 128 + nl] = acc[r];
        }
        __syncthreads();

        // ---- fused gates + state update: thread owns (s, s+8) x col jp ----
        float* hw = hgl + (tt & 1) * (16 * HID);
        {
            int s = tid >> 5;
            #pragma unroll
            for (int u = 0; u < 2; ++u, s += 8) {
                float gi = g_stage[s * 128 +       jp] + bi;
                float gf = g_stage[s * 128 +  32 + jp] + bfv;
                float gc = g_stage[s * 128 +  64 + jp] + bgg;
                float go = g_stage[s * 128 +  96 + jp] + bo;
                float cp = u ? c1 : c0;
                float cn = sigmoid_(gf) * cp + sigmoid_(gi) * tanh_(gc);
                if (u) c1 = cn; else c0 = cn;
                float hv = sigmoid_(go) * tanh_(cn);
                hw[s * HID + gcol] = hv;                           // broadcast
                hs[(size_t)((s0 + s) * TT + t) * HID + gcol] = hv; // output
            }
        }

        // ---- per-group barrier (8 WGs), monotonic counter, 1 per step ----
        if (tt + 1 < TT) {
            __threadfence();              // h stores visible at device scope
            __syncthreads();
            if (tid == 0) {
                __hip_atomic_fetch_add(ctr, 1u, __ATOMIC_RELEASE,
                                       __HIP_MEMORY_SCOPE_AGENT);
                const unsigned target = 8u * (unsigned)(tt + 1);
                while (__hip_atomic_load(ctr, __ATOMIC_ACQUIRE,
                                         __HIP_MEMORY_SCOPE_AGENT) < target)
                    __builtin_amdgcn_s_sleep(2);
            }
            __syncthreads();
        }
    }
}

// ============================================================
// Kernel 3: out[b,t,l] = [h_f|h_b](b,t,:) @ W_out[l,:]^T + b_out[l]
// grid = (T/16, B), block = 64 (2 waves); wave w -> latent cols [16w,16w+16)
// ============================================================
__global__ __launch_bounds__(64) void out_kernel(
    const float* __restrict__ hs_f, const float* __restrict__ hs_b,
    const float* __restrict__ W_out, const float* __restrict__ b_out,
    float* __restrict__ out)
{
    const int lane = threadIdx.x & 31;
    const int wave = threadIdx.x >> 5;
    const int half = lane >> 4;
    const int l16  = lane & 15;
    const int t0   = blockIdx.x * 16;
    const int b    = blockIdx.y;
    const int n    = wave * 16 + l16;

    const float* Af = hs_f + (size_t)(b * TT + t0 + l16) * HID + 2 * half;
    const float* Ab = hs_b + (size_t)(b * TT + t0 + l16) * HID + 2 * half;
    const float* Wp = W_out + (size_t)n * (2 * HID) + 2 * half;

    v8f acc = {};
    for (int k0 = 0; k0 < HID; k0 += 4) {
        v2f a  = *(const v2f*)(Af + k0);
        v2f bv = *(const v2f*)(Wp + k0);
        acc = wmma_f32(a, bv, acc);
    }
    for (int k0 = 0; k0 < HID; k0 += 4) {
        v2f a  = *(const v2f*)(Ab + k0);
        v2f bv = *(const v2f*)(Wp + HID + k0);
        acc = wmma_f32(a, bv, acc);
    }
    const float bias = b_out[n];
    #pragma unroll
    for (int r = 0; r < 8; ++r) {
        int m = r + 8 * half;
        out[(size_t)(b * TT + t0 + m) * LATENT + n] = acc[r] + bias;
    }
}

// ============================================================
extern "C" void kernel_launch(void* const* d_in, const int* in_sizes, int n_in,
                              void* d_out, int out_size, void* d_ws, size_t ws_size,
                              hipStream_t stream)
{
    const float* x     = (const float*)d_in[0];
    const int*   dsi   = (const int*)  d_in[1];
    const float* W_enc = (const float*)d_in[2];
    const float* b_enc = (const float*)d_in[3];
    const float* Wih_f = (const float*)d_in[4];
    const float* Whh_f = (const float*)d_in[5];
    const float* b_f   = (const float*)d_in[6];
    const float* Wih_b = (const float*)d_in[7];
    const float* Whh_b = (const float*)d_in[8];
    const float* b_b   = (const float*)d_in[9];
    const float* W_out = (const float*)d_in[10];
    const float* b_out = (const float*)d_in[11];
    float* out = (float*)d_out;

    // workspace: emb | hs_f | hs_b | barrier counters | h broadcast buffers
    float* ws = (float*)d_ws;
    const size_t SEG = (size_t)BB * TT * HID;
    float*    emb   = ws;
    float*    hs_f  = ws + SEG;
    float*    hs_b  = ws + 2 * SEG;
    unsigned* bars  = (unsigned*)(ws + 3 * SEG);          // 16 x 32 u32 (2 KB)
    float*    hglob = ws + 3 * SEG + 512;                 // 16 x 2 x 4096 f32

    // reset barrier counters every launch (graph-capture safe, deterministic)
    hipMemsetAsync(bars, 0, 16 * 32 * sizeof(unsigned), stream);

    dim3 g1(TT / 16, BB);
    encoder_kernel<<<g1, 256, 0, stream>>>(x, dsi, W_enc, b_enc, emb);

    dim3 g2(8, BB / 16, 2);   // 8 gate-slices x 8 sample tiles x 2 directions
    lstm_kernel<<<g2, 256, 0, stream>>>(emb, Wih_f, Whh_f, b_f,
                                        Wih_b, Whh_b, b_b, hs_f, hs_b,
                                        bars, hglob);

    dim3 g3(TT / 16, BB);
    out_kernel<<<g3, 64, 0, stream>>>(hs_f, hs_b, W_out, b_out, out);
}